// Center_TripletLoss_26517128086175
// MI455X (gfx1250) — compile-verified
//
#include <hip/hip_runtime.h>
#include <hip/hip_bf16.h>

// MI455X / gfx1250 implementation of Center_TripletLoss.
// Phase 1: segment-mean centers (HBM-bound, ~67MB @ 23.3TB/s -> ~3us, dominates).
// Phase 2: G = c1 @ c2^T via V_WMMA_F32_16X16X4_F32 (full f32 precision; GEMM is
//          only ~1 GFLOP and L2-resident, so no reason to quantize).
// Phase 3: algebraic epilogue: diag-Dmat = sq2[i]-sq2[j]-2G[ii]+2G[ij] (sq1 cancels),
//          row-max, serial cummax + sum -> scalar.

typedef __attribute__((ext_vector_type(2))) float v2f;
typedef __attribute__((ext_vector_type(8))) float v8f;

#define DIM   2048
#define NCLS  512
#define KPC   8      // samples per class (4096 / 512)

// ---------------- Phase 1: per-class centers for both modalities ----------------
__global__ void __launch_bounds__(256)
centers_kernel(const float* __restrict__ in1, const float* __restrict__ in2,
               float* __restrict__ c1, float* __restrict__ c2) {
  int idx = blockIdx.x * 256 + threadIdx.x;      // 0 .. 2*NCLS*DIM-1
  const int per = NCLS * DIM;
  const float* src = (idx < per) ? in1 : in2;
  float*       dst = (idx < per) ? c1  : c2;
  int e   = (idx < per) ? idx : idx - per;
  int cls = e >> 11;                             // e / DIM
  int d   = e & (DIM - 1);
  const float* p = src + (size_t)(cls * KPC) * DIM + d;
  float s = 0.f;
#pragma unroll
  for (int k = 0; k < KPC; ++k) s += p[(size_t)k * DIM];
  dst[e] = s * (1.0f / KPC);
}

// ---------------- Phase 2: G = c1 @ c2^T with f32 WMMA ----------------
// One wave computes one 16x16 tile of G. A-matrix (16x4 f32) lane layout:
//   lanes 0-15  : M = lane, VGPR0/1 = K = k, k+1   -> contiguous float2 load
//   lanes 16-31 : M = lane-16, VGPR0/1 = K = k+2,k+3
// B-matrix mirrors with N in place of M (B[k][n] = c2[n][k]).
__global__ void __launch_bounds__(256)
gemm_wmma_f32(const float* __restrict__ c1, const float* __restrict__ c2,
              float* __restrict__ G) {
  const int lane = threadIdx.x & 31;
  const int wave = threadIdx.x >> 5;
  const int tile = blockIdx.x * 8 + wave;        // 0..1023 (32x32 tiles)
  const int tm   = (tile >> 5) << 4;             // base row in c1
  const int tn   = (tile & 31) << 4;             // base row in c2
  const int half = lane >> 4;                    // 0: K 0,1   1: K 2,3
  const int r    = lane & 15;

  const float* aP = c1 + (size_t)(tm + r) * DIM + 2 * half;
  const float* bP = c2 + (size_t)(tn + r) * DIM + 2 * half;

  v8f acc = {};
#pragma unroll 4
  for (int k = 0; k < DIM; k += 4) {
    v2f a = *(const v2f*)(aP + k);
    v2f b = *(const v2f*)(bP + k);
    // (neg_a, A, neg_b, B, c_mod, C, reuse_a, reuse_b)
    acc = __builtin_amdgcn_wmma_f32_16x16x4_f32(false, a, false, b,
                                                (short)0, acc, false, false);
  }

  // C/D layout: VGPR v -> M = v (+8 for lanes 16-31), N = r. 16-wide coalesced stores.
  float* gRow = G + (size_t)(tm + 8 * half) * NCLS + tn + r;
#pragma unroll
  for (int v = 0; v < 8; ++v) gRow[(size_t)v * NCLS] = acc[v];
}

// ---------------- Phase 3a: sq2[j] = ||c2[j]||^2 ----------------
__global__ void __launch_bounds__(256)
sq2_kernel(const float* __restrict__ c2, float* __restrict__ sq2) {
  __shared__ float red[256];
  const float* p = c2 + (size_t)blockIdx.x * DIM;
  float s = 0.f;
  for (int d = threadIdx.x; d < DIM; d += 256) { float v = p[d]; s += v * v; }
  red[threadIdx.x] = s;
  __syncthreads();
  for (int off = 128; off > 0; off >>= 1) {
    if (threadIdx.x < off) red[threadIdx.x] += red[threadIdx.x + off];
    __syncthreads();
  }
  if (threadIdx.x == 0) sq2[blockIdx.x] = red[0];
}

// ---------------- Phase 3b: row-max, cummax, sum ----------------
// t[i][j] = diag[i]-Dmat[i][j] = sq2[i]-sq2[j]-2G[ii]+2G[ij]; t[i][i]==0 so the
// max over j already honors the max(.,0) clamp.
__global__ void __launch_bounds__(512)
finalize_kernel(const float* __restrict__ G, const float* __restrict__ sq2,
                float* __restrict__ out) {
  __shared__ float rowmax[NCLS];
  const int i = threadIdx.x;
  const float* g = G + (size_t)i * NCLS;
  float m = -3.0e38f;
  for (int j = 0; j < NCLS; ++j) m = fmaxf(m, 2.0f * g[j] - sq2[j]);
  rowmax[i] = sq2[i] - 2.0f * g[i] + m;
  __syncthreads();
  if (i == 0) {
    float run = -3.0e38f, s = 0.f;
    for (int t = 0; t < NCLS; ++t) { run = fmaxf(run, rowmax[t]); s += run; }
    out[0] = s;
  }
}

extern "C" void kernel_launch(void* const* d_in, const int* in_sizes, int n_in,
                              void* d_out, int out_size, void* d_ws, size_t ws_size,
                              hipStream_t stream) {
  const float* in1 = (const float*)d_in[0];
  const float* in2 = (const float*)d_in[1];
  float* out = (float*)d_out;

  // Workspace layout (floats): c1 | c2 | G | sq2  (~9.4 MB total)
  float* c1  = (float*)d_ws;
  float* c2  = c1 + (size_t)NCLS * DIM;
  float* G   = c2 + (size_t)NCLS * DIM;
  float* sq2 = G  + (size_t)NCLS * NCLS;

  centers_kernel<<<(2 * NCLS * DIM) / 256, 256, 0, stream>>>(in1, in2, c1, c2);
  sq2_kernel<<<NCLS, 256, 0, stream>>>(c2, sq2);
  gemm_wmma_f32<<<(NCLS / 16) * (NCLS / 16) / 8, 256, 0, stream>>>(c1, c2, G);
  finalize_kernel<<<1, NCLS, 0, stream>>>(G, sq2, out);
}